// QueryAndGroup_17214228923002
// MI455X (gfx1250) — compile-verified
//
#include <hip/hip_runtime.h>
#include <stdint.h>

typedef __attribute__((ext_vector_type(2))) float v2f;
typedef __attribute__((ext_vector_type(8))) float v8f;

// Problem constants (match reference setup_inputs)
constexpr int B_ = 2;
constexpr int N_ = 16384;
constexpr int M_ = 4096;
constexpr int E_ = 32;
constexpr int C_ = 64;
constexpr int S_ = 32;            // NSAMPLE
constexpr int SEG_ = 8;           // N split into 8 ordered segments
constexpr int SEGLEN_ = N_ / SEG_;// 2048 points per segment (24 KB in LDS)
constexpr float R2 = 0.1f * 0.1f;

constexpr int TILES_ = M_ / 16;            // 256 center tiles per batch
constexpr int WAVES_PER_BLOCK_ = 8;        // 256 threads
constexpr int BLOCKS_PER_SEG_ = B_ * TILES_ / WAVES_PER_BLOCK_;  // 64
constexpr int TILEBLK_ = TILES_ / WAVES_PER_BLOCK_;              // 32

// d_ws layout (ints):
//   [0, IDX_INTS)                 final idx  (B,M,S)        1 MB
//   [IDX_INTS, +CNT_INTS)         seg counts (B,M,SEG)      256 KB
//   [.., +SEGIDX_INTS)            seg hits   (B,M,SEG,S)    8 MB
constexpr size_t IDX_INTS = (size_t)B_ * M_ * S_;
constexpr size_t CNT_INTS = (size_t)B_ * M_ * SEG_;

// ---------------------------------------------------------------------------
// Kernel 1: segmented ball query via V_WMMA_F32_16X16X4_F32, points staged
// in LDS. One block = 8 waves x 16 centers, all sharing one 2048-point
// segment (copied once, coalesced float4). Inner loop reads points from LDS
// (short-latency ds loads), 32 points per iteration = 2 WMMAs.
//   A (16x4, pre-scaled by -2): lane l -> center M=l%16, K striped by half
//   B (4x16):                   lane l -> point  N=l%16, K striped by half
//   C seeded with |c|^2+|p|^2   -> D = squared distances directly.
// D layout: lane l, vgpr v -> (M = v + 8*(l/16), N = l%16): ballot(d[v]<r2)
// carries center v in bits 0..15 and center v+8 in bits 16..31.
// Owner lanes 0..15 append ascending hit indices for their center.
// ---------------------------------------------------------------------------
__global__ __launch_bounds__(256) void ball_query_wmma_seg(
    const float* __restrict__ xyz,      // (B,N,3)
    const float* __restrict__ new_xyz,  // (B,M,3)
    int* __restrict__ seg_idx,          // (B,M,SEG,S)
    int* __restrict__ seg_cnt)          // (B,M,SEG)
{
  __shared__ float spts[SEGLEN_ * 3];   // 24 KB of 320 KB/WGP

  const int tid  = (int)threadIdx.x;
  const int lane = tid & 31;
  const int widx = tid >> 5;

  // Block-level mapping: every wave of this block shares (seg, b).
  const int seg = (int)blockIdx.x / BLOCKS_PER_SEG_;
  const int rb  = (int)blockIdx.x % BLOCKS_PER_SEG_;
  const int b   = rb / TILEBLK_;
  const int m0  = ((rb % TILEBLK_) * WAVES_PER_BLOCK_ + widx) * 16;

  const int nbeg = seg * SEGLEN_;

  // Cooperative stage of the segment into LDS: 6 float4 per thread.
  {
    const float4* __restrict__ src = reinterpret_cast<const float4*>(
        xyz + (size_t)b * N_ * 3 + (size_t)nbeg * 3);   // 16B aligned
    float4* __restrict__ dst = reinterpret_cast<float4*>(spts);
#pragma unroll
    for (int k = 0; k < (SEGLEN_ * 3 / 4) / 256; ++k)   // 1536/256 = 6
      dst[tid + k * 256] = src[tid + k * 256];
  }
  __syncthreads();   // LDS read-only after this; waves may diverge freely.

  const int half = lane >> 4;
  const int li   = lane & 15;

  // Load this lane's center (both halves load the same 16 centers; cached).
  const float* cp = new_xyz + ((size_t)b * M_ + (size_t)(m0 + li)) * 3;
  const float cx = cp[0], cy = cp[1], cz = cp[2];
  const float cn_lane = cx * cx + cy * cy + cz * cz;

  // A matrix, pre-scaled by -2 so the WMMA product is -2*c.p
  v2f a;
  a.x = half ? (-2.0f * cz) : (-2.0f * cx);
  a.y = half ? 0.0f         : (-2.0f * cy);

  // Per-lane copy of |c|^2 for the 8 center rows this lane contributes to.
  float cn[8];
#pragma unroll
  for (int v = 0; v < 8; ++v)
    cn[v] = __shfl(cn_lane, v + 8 * half, 32);

  int count = 0;
  const size_t cbase = (size_t)b * M_ + (size_t)(m0 + li); // in-range for all lanes
  int* __restrict__ myout = seg_idx + (cbase * SEG_ + (size_t)seg) * S_;

  // One 16-point chunk: C-seed, WMMA, mask routing, ordered append.
  auto process = [&](float qx, float qy, float qz, int n0) {
    const float pn = qx * qx + qy * qy + qz * qz;
    v2f bm;
    bm.x = half ? qz   : qx;
    bm.y = half ? 0.0f : qy;

    v8f c;
#pragma unroll
    for (int v = 0; v < 8; ++v) c[v] = cn[v] + pn;

    // D = (-2A) * B + (|c|^2+|p|^2)  ==  squared distances, 16x16 tile.
    v8f d = __builtin_amdgcn_wmma_f32_16x16x4_f32(
        false, a, false, bm, (short)0, c, false, false);

    uint32_t mymask = 0;
#pragma unroll
    for (int v = 0; v < 8; ++v) {
      const uint32_t bl = (uint32_t)__ballot(d[v] < R2);
      if (lane == v)     mymask = bl & 0xFFFFu;
      if (lane == v + 8) mymask = bl >> 16;
    }

    while (mymask && count < S_) {
      const int bit = __builtin_ctz(mymask);
      mymask &= mymask - 1;
      myout[count++] = n0 + bit;
    }
  };

  // 64 iterations of 32 points, all point reads from LDS.
  for (int n0r = 0; n0r < SEGLEN_; n0r += 32) {
    const float* pA = spts + (size_t)(n0r + li) * 3;
    const float ax = pA[0], ay = pA[1], az = pA[2];
    const float* pB = spts + (size_t)(n0r + 16 + li) * 3;
    const float bx = pB[0], by = pB[1], bz = pB[2];

    process(ax, ay, az, nbeg + n0r);
    process(bx, by, bz, nbeg + n0r + 16);

    // Uniform (per-wave) early exit once all 16 centers are full here.
    if ((((uint32_t)__ballot(count < S_)) & 0xFFFFu) == 0u) break;
  }

  if (lane < 16) seg_cnt[cbase * SEG_ + (size_t)seg] = count;
}

// ---------------------------------------------------------------------------
// Kernel 2: merge per-segment ordered hit lists into final idx (B,M,S),
// applying pointnet2 padding (first valid index, or 0 if none).
// One thread per center; small, L2-resident traffic.
// ---------------------------------------------------------------------------
__global__ __launch_bounds__(256) void merge_segments(
    const int* __restrict__ seg_idx,  // (B,M,SEG,S)
    const int* __restrict__ seg_cnt,  // (B,M,SEG)
    int* __restrict__ ws_idx)         // (B,M,S)
{
  const int c = (int)(blockIdx.x * blockDim.x + threadIdx.x);
  if (c >= B_ * M_) return;
  int* __restrict__ out = ws_idx + (size_t)c * S_;
  int cnt = 0, first = 0;
  for (int seg = 0; seg < SEG_ && cnt < S_; ++seg) {
    const int sc = seg_cnt[(size_t)c * SEG_ + seg];
    const int* p = seg_idx + ((size_t)c * SEG_ + seg) * S_;
    for (int k = 0; k < sc && cnt < S_; ++k) {
      const int n = p[k];
      if (cnt == 0) first = n;
      out[cnt++] = n;
    }
  }
  const int pad = cnt ? first : 0;
  for (int s = cnt; s < S_; ++s) out[s] = pad;
}

// ---------------------------------------------------------------------------
// Kernel 3: grouping/gather. Output (B, E+C, M, S) f32, float4 per thread
// (coalesced 128-bit stores). Gathers hit L2 (all inputs << 192 MB).
// ---------------------------------------------------------------------------
__global__ __launch_bounds__(256) void group_gather(
    const float* __restrict__ xyz_embed,      // (B,N,E)
    const float* __restrict__ new_xyz_embed,  // (B,M,E)
    const float* __restrict__ features,       // (B,C,N)
    const int* __restrict__ ws_idx,           // (B,M,S)
    float* __restrict__ out)                  // (B,E+C,M,S)
{
  const int total4 = B_ * (E_ + C_) * M_ * S_ / 4;
  const int i = (int)(blockIdx.x * blockDim.x + threadIdx.x);
  if (i >= total4) return;

  const int q  = i & 7;          // which int4 of the 32 samples
  const int t  = i >> 3;         // (b*(E+C)+ch)*M + m
  const int m  = t & (M_ - 1);   // M = 4096 = 2^12
  const int u  = t >> 12;        // b*(E+C) + ch
  const int ch = u % (E_ + C_);
  const int b  = u / (E_ + C_);

  const int4 idx4 =
      reinterpret_cast<const int4*>(ws_idx)[((b * M_ + m) << 3) + q];

  float4 v;
  if (ch < E_) {
    const float sub = new_xyz_embed[((b * M_ + m) << 5) + ch]; // E_=32
    const float* eb = xyz_embed + (b * N_ * E_) + ch;
    v.x = eb[idx4.x * E_] - sub;
    v.y = eb[idx4.y * E_] - sub;
    v.z = eb[idx4.z * E_] - sub;
    v.w = eb[idx4.w * E_] - sub;
  } else {
    const float* fb = features + (b * C_ + (ch - E_)) * N_;
    v.x = fb[idx4.x];
    v.y = fb[idx4.y];
    v.z = fb[idx4.z];
    v.w = fb[idx4.w];
  }
  reinterpret_cast<float4*>(out)[i] = v;
}

// ---------------------------------------------------------------------------
extern "C" void kernel_launch(void* const* d_in, const int* in_sizes, int n_in,
                              void* d_out, int out_size, void* d_ws, size_t ws_size,
                              hipStream_t stream) {
  (void)in_sizes; (void)n_in; (void)out_size; (void)ws_size;
  const float* xyz           = (const float*)d_in[0];
  const float* xyz_embed     = (const float*)d_in[1];
  const float* new_xyz       = (const float*)d_in[2];
  const float* new_xyz_embed = (const float*)d_in[3];
  const float* features      = (const float*)d_in[4];
  float* out = (float*)d_out;

  int* ws_idx  = (int*)d_ws;            // (B,M,S)        final indices
  int* seg_cnt = ws_idx + IDX_INTS;     // (B,M,SEG)      per-segment counts
  int* seg_idx = seg_cnt + CNT_INTS;    // (B,M,SEG,S)    per-segment hits

  // 1) segmented ball query: 8 segments x 64 blocks = 512 blocks (4096 waves)
  const int blocks1 = SEG_ * BLOCKS_PER_SEG_;
  ball_query_wmma_seg<<<blocks1, 256, 0, stream>>>(xyz, new_xyz,
                                                   seg_idx, seg_cnt);

  // 2) ordered merge of segment hit lists
  const int blocks2 = (B_ * M_ + 255) / 256;            // 32 blocks
  merge_segments<<<blocks2, 256, 0, stream>>>(seg_idx, seg_cnt, ws_idx);

  // 3) gather + subtract + concat (bandwidth-bound, float4 stores)
  const int total4  = B_ * (E_ + C_) * M_ * S_ / 4;
  const int blocks3 = (total4 + 255) / 256;             // 24576 blocks
  group_gather<<<blocks3, 256, 0, stream>>>(xyz_embed, new_xyz_embed,
                                            features, ws_idx, out);
}